// SDF_43533788512448
// MI455X (gfx1250) — compile-verified
//
#include <hip/hip_runtime.h>
#include <math.h>
#include <float.h>

// CDNA5 wave32 WMMA types: A/B for V_WMMA_F32_16X16X4_F32 are 64 f32 / 32 lanes = v2f,
// C/D accumulator 16x16 f32 = 256 f32 / 32 lanes = v8f.
typedef __attribute__((ext_vector_type(2))) float v2f;
typedef __attribute__((ext_vector_type(8))) float v8f;

#define EPSF 1e-12f

__device__ __forceinline__ float safef(float x) {
    return (fabsf(x) < EPSF) ? EPSF : x;
}

// Fast reciprocal (v_rcp_f32, ~1 ulp of 2^-23 relative): used only in pass 1,
// where distances are only *ranked*, never emitted. Avoids the IEEE
// div_scale/div_fmas/div_fixup expansion (~10 VALU ops per divide).
__device__ __forceinline__ float fastrcp(float x) {
    return __builtin_amdgcn_rcpf(x);
}

// ---------------------------------------------------------------------------
// Kernel 1: per-face precompute.
// For face f with vertices a,b,c: build three WMMA "B columns" (4-vectors that
// dot with (px,py,pz,1)) and four scalar constants.
//   col1 = (ab, -ab.a)      -> d1 = ab.(p-a)
//   col2 = (ac, -ac.a)      -> d2 = ac.(p-a)
//   col3 = (-2a, a.a)       -> e  = |p-a|^2 - |p|^2   (|p|^2 const per query)
//   consts = (|ab|^2, |ac|^2, ab.ac, |bc|^2)
// Layout: 16 floats per face, 64B aligned.
// ---------------------------------------------------------------------------
__global__ void sdf_prep(const int* __restrict__ faces,
                         const float* __restrict__ verts,
                         float* __restrict__ fdat, int nF, int nFpad) {
    int f = blockIdx.x * blockDim.x + threadIdx.x;
    if (f >= nFpad) return;
    float* o = fdat + (size_t)f * 16;
    if (f >= nF) {
#pragma unroll
        for (int i = 0; i < 16; ++i) o[i] = 0.0f;   // padded faces: masked in pass1
        return;
    }
    int ia = faces[3 * f + 0] * 3, ib = faces[3 * f + 1] * 3, ic = faces[3 * f + 2] * 3;
    float ax = verts[ia], ay = verts[ia + 1], az = verts[ia + 2];
    float bx = verts[ib], by = verts[ib + 1], bz = verts[ib + 2];
    float cx = verts[ic], cy = verts[ic + 1], cz = verts[ic + 2];
    float abx = bx - ax, aby = by - ay, abz = bz - az;
    float acx = cx - ax, acy = cy - ay, acz = cz - az;
    float ab_a  = abx * ax + aby * ay + abz * az;
    float ac_a  = acx * ax + acy * ay + acz * az;
    float aa    = ax * ax + ay * ay + az * az;
    float Kabab = abx * abx + aby * aby + abz * abz;
    float Kacac = acx * acx + acy * acy + acz * acz;
    float Kabac = abx * acx + aby * acy + abz * acz;
    float Kbcbc = Kabab + Kacac - 2.0f * Kabac;
    o[0]  = abx;        o[1]  = aby;        o[2]  = abz;        o[3]  = -ab_a;
    o[4]  = acx;        o[5]  = acy;        o[6]  = acz;        o[7]  = -ac_a;
    o[8]  = -2.0f * ax; o[9]  = -2.0f * ay; o[10] = -2.0f * az; o[11] = aa;
    o[12] = Kabab;      o[13] = Kacac;      o[14] = Kabac;      o[15] = Kbcbc;
}

// ---------------------------------------------------------------------------
// Kernel 2: pass 1 — WMMA brute-force argmin over faces.
// One wave owns a 16-query tile. Loop over 16-face tiles:
//   D1/D2/E = P(16x4) x B(4x16) via v_wmma_f32_16x16x4_f32 (3 WMMAs/tile).
// Then per output element run the Ericson region selection on (d1,d2,e) +
// face constants (priority identical to the reference's where-override order)
// to get squared distance minus |p|^2, and track min+argmin.
// C/D layout: VGPR j, lane L -> (M = j + (L>=16 ? 8 : 0), N = L & 15).
// ---------------------------------------------------------------------------
__global__ void __launch_bounds__(256)
sdf_pass1(const float* __restrict__ query,
          const float* __restrict__ fdat,
          int* __restrict__ bestIdx, int nQ, int nF, int nTilesF) {
    const int lane = threadIdx.x & 31;
    const int wave = threadIdx.x >> 5;
    const int qtile = blockIdx.x * (blockDim.x >> 5) + wave;
    const int qbase = qtile * 16;
    if (qbase >= nQ) return;                       // uniform per wave

    const bool hi = lane >= 16;
    const int  M  = lane & 15;
    int qi = qbase + M;
    if (qi >= nQ) qi = nQ - 1;                     // clamp; dup results discarded

    const float px = query[3 * qi], py = query[3 * qi + 1], pz = query[3 * qi + 2];
    // A-matrix 16x4 layout: lanes 0-15 hold K0(v0),K1(v1); lanes 16-31 hold K2,K3.
    v2f A;
    A.x = hi ? pz : px;
    A.y = hi ? 1.0f : py;

    float minv[8];
    int   mini[8];
#pragma unroll
    for (int j = 0; j < 8; ++j) { minv[j] = FLT_MAX; mini[j] = 0; }

    const int N = lane & 15;
    const float4* __restrict__ fd4 = reinterpret_cast<const float4*>(fdat);

    for (int t = 0; t < nTilesF; ++t) {
        const int f = t * 16 + N;                  // this lane's face column
        float4 c1 = fd4[f * 4 + 0];
        float4 c2 = fd4[f * 4 + 1];
        float4 c3 = fd4[f * 4 + 2];
        float4 cc = fd4[f * 4 + 3];

        // B-matrix 4x16 layout: lanes 0-15 -> (K0,K1), lanes 16-31 -> (K2,K3), N = lane&15.
        v2f B1, B2, B3;
        B1.x = hi ? c1.z : c1.x;  B1.y = hi ? c1.w : c1.y;
        B2.x = hi ? c2.z : c2.x;  B2.y = hi ? c2.w : c2.y;
        B3.x = hi ? c3.z : c3.x;  B3.y = hi ? c3.w : c3.y;

        v8f z = {};
        v8f D1 = __builtin_amdgcn_wmma_f32_16x16x4_f32(false, A, false, B1, (short)0, z, false, false);
        v8f D2 = __builtin_amdgcn_wmma_f32_16x16x4_f32(false, A, false, B2, (short)0, z, false, false);
        v8f E  = __builtin_amdgcn_wmma_f32_16x16x4_f32(false, A, false, B3, (short)0, z, false, false);

        const bool valid = f < nF;
        const float Kabab = cc.x, Kacac = cc.y, Kabac = cc.z, Kbcbc = cc.w;

#pragma unroll
        for (int j = 0; j < 8; ++j) {
            float d1 = D1[j], d2 = D2[j], e = E[j];
            float d3 = d1 - Kabab;
            float d4 = d2 - Kabac;
            float d5 = d1 - Kabac;
            float d6 = d2 - Kacac;
            float vc = d1 * d4 - d3 * d2;
            float vb = d5 * d2 - d1 * d6;
            float va = d3 * d6 - d5 * d4;

            // interior (default): one fast rcp shared by v and w
            float rdn = fastrcp(safef(va + vb + vc));
            float v = vb * rdn, w = vc * rdn;
            float rel = e - 2.0f * v * d1 - 2.0f * w * d2
                        + v * v * Kabab + w * w * Kacac + 2.0f * v * w * Kabac;

            float u43 = d4 - d3, u56 = d5 - d6;
            if (va <= 0.0f && u43 >= 0.0f && u56 >= 0.0f) {        // edge BC
                float tt = u43 * fastrcp(safef(u43 + u56));
                rel = (e - 2.0f * d1 + Kabab) - tt * (2.0f * u43 - tt * Kbcbc);
            }
            if (vb <= 0.0f && d2 >= 0.0f && d6 <= 0.0f) {          // edge AC
                float tt = d2 * fastrcp(safef(d2 - d6));
                rel = e - tt * (2.0f * d2 - tt * Kacac);
            }
            if (d6 >= 0.0f && d5 <= d6)                            // vertex C
                rel = e - 2.0f * d2 + Kacac;
            if (vc <= 0.0f && d1 >= 0.0f && d3 <= 0.0f) {          // edge AB
                float tt = d1 * fastrcp(safef(d1 - d3));
                rel = e - tt * (2.0f * d1 - tt * Kabab);
            }
            if (d3 >= 0.0f && d4 <= d3)                            // vertex B
                rel = e - 2.0f * d1 + Kabab;
            if (d1 <= 0.0f && d2 <= 0.0f)                          // vertex A
                rel = e;

            if (!valid) rel = FLT_MAX;
            if (rel < minv[j]) { minv[j] = rel; mini[j] = f; }
        }
    }

    // Reduce across the 16 face-columns (lanes within each half-wave).
#pragma unroll
    for (int j = 0; j < 8; ++j) {
        float v = minv[j];
        int   i = mini[j];
        for (int m = 8; m > 0; m >>= 1) {
            float ov = __shfl_xor(v, m, 16);
            int   oi = __shfl_xor(i, m, 16);
            if (ov < v || (ov == v && oi < i)) { v = ov; i = oi; }
        }
        minv[j] = v; mini[j] = i;
    }

    if ((lane & 15) == 0) {
#pragma unroll
        for (int j = 0; j < 8; ++j) {
            int q = qbase + j + (hi ? 8 : 0);
            if (q < nQ) bestIdx[q] = mini[j];
        }
    }
}

// ---------------------------------------------------------------------------
// Kernel 3: pass 2 — exact reference recomputation for the winning face.
// Reproduces the reference's Ericson code and where-override order exactly
// (IEEE-accurate division kept here on purpose), then cp, squared distance,
// sign via un-normalized normal, sdist.  out[q] = (cp.x, cp.y, cp.z, sdist)
// ---------------------------------------------------------------------------
__global__ void sdf_pass2(const float* __restrict__ query,
                          const float* __restrict__ verts,
                          const int* __restrict__ faces,
                          const int* __restrict__ bestIdx,
                          float* __restrict__ out, int nQ) {
    int q = blockIdx.x * blockDim.x + threadIdx.x;
    if (q >= nQ) return;
    int f = bestIdx[q];
    int ia = faces[3 * f + 0] * 3, ib = faces[3 * f + 1] * 3, ic = faces[3 * f + 2] * 3;
    float ax = verts[ia], ay = verts[ia + 1], az = verts[ia + 2];
    float bx = verts[ib], by = verts[ib + 1], bz = verts[ib + 2];
    float cx = verts[ic], cy = verts[ic + 1], cz = verts[ic + 2];
    float px = query[3 * q], py = query[3 * q + 1], pz = query[3 * q + 2];

    float abx = bx - ax, aby = by - ay, abz = bz - az;
    float acx = cx - ax, acy = cy - ay, acz = cz - az;
    float apx = px - ax, apy = py - ay, apz = pz - az;
    float d1 = abx * apx + aby * apy + abz * apz;
    float d2 = acx * apx + acy * apy + acz * apz;
    float bpx = px - bx, bpy = py - by, bpz = pz - bz;
    float d3 = abx * bpx + aby * bpy + abz * bpz;
    float d4 = acx * bpx + acy * bpy + acz * bpz;
    float cpx = px - cx, cpy = py - cy, cpz = pz - cz;
    float d5 = abx * cpx + aby * cpy + abz * cpz;
    float d6 = acx * cpx + acy * cpy + acz * cpz;
    float vc = d1 * d4 - d3 * d2;
    float vb = d5 * d2 - d1 * d6;
    float va = d3 * d6 - d5 * d4;
    float t_ab = d1 / safef(d1 - d3);
    float t_ac = d2 / safef(d2 - d6);
    float t_bc = (d4 - d3) / safef((d4 - d3) + (d5 - d6));
    float dn = safef(va + vb + vc);
    float v = vb / dn, w = vc / dn;

    float rx = ax + abx * v + acx * w;
    float ry = ay + aby * v + acy * w;
    float rz = az + abz * v + acz * w;
    if (va <= 0.0f && (d4 - d3) >= 0.0f && (d5 - d6) >= 0.0f) {
        rx = bx + (cx - bx) * t_bc; ry = by + (cy - by) * t_bc; rz = bz + (cz - bz) * t_bc;
    }
    if (vb <= 0.0f && d2 >= 0.0f && d6 <= 0.0f) {
        rx = ax + acx * t_ac; ry = ay + acy * t_ac; rz = az + acz * t_ac;
    }
    if (d6 >= 0.0f && d5 <= d6) { rx = cx; ry = cy; rz = cz; }
    if (vc <= 0.0f && d1 >= 0.0f && d3 <= 0.0f) {
        rx = ax + abx * t_ab; ry = ay + aby * t_ab; rz = az + abz * t_ab;
    }
    if (d3 >= 0.0f && d4 <= d3) { rx = bx; ry = by; rz = bz; }
    if (d1 <= 0.0f && d2 <= 0.0f) { rx = ax; ry = ay; rz = az; }

    float dxp = px - rx, dyp = py - ry, dzp = pz - rz;
    float dist2 = dxp * dxp + dyp * dyp + dzp * dzp;
    float nx = aby * acz - abz * acy;
    float ny = abz * acx - abx * acz;
    float nz = abx * acy - aby * acx;
    float sgn = (dxp * nx + dyp * ny + dzp * nz) >= 0.0f ? 1.0f : -1.0f;
    float sd = sgn * sqrtf(fmaxf(dist2, 0.0f));
    out[4 * q + 0] = rx;
    out[4 * q + 1] = ry;
    out[4 * q + 2] = rz;
    out[4 * q + 3] = sd;
}

// ---------------------------------------------------------------------------
extern "C" void kernel_launch(void* const* d_in, const int* in_sizes, int n_in,
                              void* d_out, int out_size, void* d_ws, size_t ws_size,
                              hipStream_t stream) {
    const int*   faces = (const int*)d_in[0];
    const float* verts = (const float*)d_in[1];
    const float* query = (const float*)d_in[2];
    float* out = (float*)d_out;

    const int nF = in_sizes[0] / 3;
    const int nQ = in_sizes[2] / 3;
    const int nFpad = (nF + 15) & ~15;

    float* fdat = (float*)d_ws;                                        // nFpad*16 floats
    int* bestIdx = (int*)((char*)d_ws + (size_t)nFpad * 16 * sizeof(float)); // nQ ints

    int prepBlocks = (nFpad + 255) / 256;
    sdf_prep<<<prepBlocks, 256, 0, stream>>>(faces, verts, fdat, nF, nFpad);

    const int nQt = (nQ + 15) / 16;          // 16-query tiles (one wave each)
    const int wavesPerBlock = 8;             // 256 threads = 8 wave32
    int p1Blocks = (nQt + wavesPerBlock - 1) / wavesPerBlock;
    sdf_pass1<<<p1Blocks, wavesPerBlock * 32, 0, stream>>>(query, fdat, bestIdx,
                                                           nQ, nF, nFpad / 16);

    int p2Blocks = (nQ + 255) / 256;
    sdf_pass2<<<p2Blocks, 256, 0, stream>>>(query, verts, faces, bestIdx, out, nQ);
}